// FeedbackModelEnb_74311524155795
// MI455X (gfx1250) — compile-verified
//
#include <hip/hip_runtime.h>
#include <hip/hip_bf16.h>
#include <math.h>

// Problem constants (match reference)
#define B_        16
#define L_        2048
#define D_        1024
#define S_        32
#define FEAT      1040          // D + 8 (width emb) + 8 (length emb); 1040 = 65*16 = 260*4
#define NLAB      3
#define NBUCKETS  64
#define WIDTH_BK  16
#define LEN_BK    32

typedef __attribute__((ext_vector_type(2))) float v2f;
typedef __attribute__((ext_vector_type(8))) float v8f;

// ---------------------------------------------------------------------------
// Kernel 1: per-example valid lengths = sum(attention_mask[b, :])
// ---------------------------------------------------------------------------
__global__ void lengths_kernel(const int* __restrict__ mask, int* __restrict__ lengths) {
    __shared__ int red[256];
    const int b = blockIdx.x;
    const int t = threadIdx.x;
    const int* row = mask + (size_t)b * L_;
    int acc = 0;
    for (int i = t; i < L_; i += 256) acc += row[i];
    red[t] = acc;
    __syncthreads();
    for (int off = 128; off > 0; off >>= 1) {
        if (t < off) red[t] += red[t + off];
        __syncthreads();
    }
    if (t == 0) lengths[b] = red[0];
}

// ---------------------------------------------------------------------------
// Kernel 2: span mean pooling + bucket-embedding concat -> feat [B*S, FEAT]
// One block per (b,s). 256 threads x float4 => each row read is one coalesced
// 4KB burst; streams the whole 134MB encoder tensor once (HBM-bound, ~6us:
// this pass IS the kernel's runtime on MI455X at 23.3 TB/s).
// ---------------------------------------------------------------------------
__global__ void span_pool_kernel(const float* __restrict__ enc,
                                 const int*   __restrict__ head,
                                 const int*   __restrict__ tail,
                                 const float* __restrict__ width_emb,
                                 const float* __restrict__ length_emb,
                                 const int*   __restrict__ lengths,
                                 float*       __restrict__ feat) {
    const int bs = blockIdx.x;
    const int b  = bs / S_;
    const int t  = threadIdx.x;

    const int h  = head[bs];
    const int tl = tail[bs];
    int start = h + 1; if (start < 0) start = 0;
    int end   = tl;    if (end > L_)  end = L_;
    int count = end - start; if (count < 0) count = 0;
    const float inv = 1.0f / (float)(count > 0 ? count : 1);

    const float* base = enc + (size_t)b * L_ * D_ + (size_t)t * 4;
    float ax = 0.f, ay = 0.f, az = 0.f, aw = 0.f;
    for (int r = start; r < end; ++r) {
        const float4 v = *(const float4*)(base + (size_t)r * D_);
        if (r + 4 < end) __builtin_prefetch(base + (size_t)(r + 4) * D_, 0, 0);
        ax += v.x; ay += v.y; az += v.z; aw += v.w;
    }
    float* frow = feat + (size_t)bs * FEAT;
    ((float4*)frow)[t] = make_float4(ax * inv, ay * inv, az * inv, aw * inv);

    // Append width/length bucket embeddings (16 extra features)
    if (t < 8) {
        int wb = (tl - h) / WIDTH_BK;
        if (wb < 0) wb = 0; if (wb > NBUCKETS - 1) wb = NBUCKETS - 1;
        frow[D_ + t] = width_emb[wb * 8 + t];
    } else if (t < 16) {
        int lb = lengths[b] / LEN_BK;
        if (lb < 0) lb = 0; if (lb > NBUCKETS - 1) lb = NBUCKETS - 1;
        frow[D_ + 8 + (t - 8)] = length_emb[lb * 8 + (t - 8)];
    }
}

// ---------------------------------------------------------------------------
// Kernel 3: logits = feat @ cls_w + cls_b via V_WMMA_F32_16X16X4_F32.
// One wave (32 threads, wave32) per 16-row output tile; N padded 3 -> 16.
//
// A tile (16 x 1040 f32 = 65 KB, contiguous rows of feat) is staged into LDS
// with the CDNA5 async copy path (global_load_async_to_lds_b128, ASYNCcnt),
// then the K loop (260 steps of K=4) is branch-free:
//   1x ds_load_b64 (A frag) + 2x cached global_load_b32 * mask (B frag) + wmma.
//
// f32 WMMA operand layout assumptions (ISA 7.12.2):
//   A 16x4 : lanes 0-15 -> K = {k, k+1}; lanes 16-31 -> K = {k+2, k+3}; M = lane&15
//   B 4x16 : VGPRv: lanes 0-15 -> row k+2*half+v, etc.; N = lane&15
//   C 16x16: VGPR v: lanes 0-15 -> M=v, lanes 16-31 -> M=v+8, N = lane&15
// ---------------------------------------------------------------------------
__global__ void gemm_wmma_kernel(const float* __restrict__ feat,
                                 const float* __restrict__ cls_w,
                                 const float* __restrict__ cls_b,
                                 float*       __restrict__ logits) {
    __shared__ float a_tile[16 * FEAT];      // 66,560 B of 320 KB LDS

    const int lane = threadIdx.x;            // 0..31 (one wave)
    const int tile = blockIdx.x;             // 0..31 -> rows [tile*16, tile*16+16)
    const int half = lane >> 4;              // 0 or 1
    const int m    = lane & 15;

    // ---- async-stage the contiguous 16 x FEAT A tile into LDS ----
    {
        const unsigned n128       = (16 * FEAT * 4) / 16;   // 4160 x 16B chunks
        const char*    gsrc       = (const char*)(feat + (size_t)tile * 16 * FEAT);
        const unsigned lds_base   = (unsigned)(size_t)&a_tile[0];
        for (unsigned i = lane; i < n128; i += 32) {
            const unsigned long long gaddr = (unsigned long long)(gsrc + (size_t)i * 16);
            const unsigned lds_off = lds_base + i * 16;
            asm volatile("global_load_async_to_lds_b128 %0, %1, off"
                         :: "v"(lds_off), "v"(gaddr) : "memory");
        }
#if __has_builtin(__builtin_amdgcn_s_wait_asynccnt)
        __builtin_amdgcn_s_wait_asynccnt(0);
#else
        asm volatile("s_wait_asynccnt 0" ::: "memory");
#endif
        __syncthreads();                     // single-wave WG: barrier is a NOP
    }

    // Branch-free B fragment: every lane loads a valid cls_w element (L0/L2
    // cached, 12 KB) and multiplies by a 0/1 column mask. No EXEC juggling.
    const int   mc   = (m < NLAB) ? m : (NLAB - 1);
    const float msel = (m < NLAB) ? 1.0f : 0.0f;
    const float* arow_lds = a_tile + (size_t)m * FEAT;

    v8f c = {0.f, 0.f, 0.f, 0.f, 0.f, 0.f, 0.f, 0.f};

#pragma unroll 4
    for (int k = 0; k < FEAT; k += 4) {
        const int kk = k + 2 * half;

        const float2 a2 = *(const float2*)(arow_lds + kk);   // ds_load_b64
        v2f a; a.x = a2.x; a.y = a2.y;

        v2f bm;
        bm.x = cls_w[kk * NLAB + mc] * msel;
        bm.y = cls_w[(kk + 1) * NLAB + mc] * msel;

        // 8 args: (neg_a, A, neg_b, B, c_mod, C, reuse_a, reuse_b)
        c = __builtin_amdgcn_wmma_f32_16x16x4_f32(
                false, a, false, bm, (short)0, c, false, false);
    }

    if (m < NLAB) {
        const float bias = cls_b[m];
#pragma unroll
        for (int v = 0; v < 8; ++v) {
            const int row = tile * 16 + v + 8 * half;
            logits[row * NLAB + m] = c[v] + bias;
        }
    }
}

// ---------------------------------------------------------------------------
// Kernel 4: cross-entropy (log_softmax + NLL, ignore_index=-1), block reduce.
// ---------------------------------------------------------------------------
__global__ void loss_kernel(const float* __restrict__ logits,
                            const int*   __restrict__ labels,
                            float*       __restrict__ out_loss) {
    __shared__ float snll[512];
    __shared__ int   scnt[512];
    const int i = threadIdx.x;               // 0..511 == B*S rows

    const float x0 = logits[i * NLAB + 0];
    const float x1 = logits[i * NLAB + 1];
    const float x2 = logits[i * NLAB + 2];
    const int lab  = labels[i];

    const float mx  = fmaxf(x0, fmaxf(x1, x2));
    const float lse = mx + logf(expf(x0 - mx) + expf(x1 - mx) + expf(x2 - mx));
    int lc = lab; if (lc < 0) lc = 0; if (lc > 2) lc = 2;
    const float sel = (lc == 0) ? x0 : ((lc == 1) ? x1 : x2);
    const bool valid = lab > -1;

    snll[i] = valid ? (lse - sel) : 0.0f;
    scnt[i] = valid ? 1 : 0;
    __syncthreads();
    for (int off = 256; off > 0; off >>= 1) {
        if (i < off) { snll[i] += snll[i + off]; scnt[i] += scnt[i + off]; }
        __syncthreads();
    }
    if (i == 0) {
        const int cnt = scnt[0] > 0 ? scnt[0] : 1;
        out_loss[0] = snll[0] / (float)cnt;
    }
}

// ---------------------------------------------------------------------------
extern "C" void kernel_launch(void* const* d_in, const int* in_sizes, int n_in,
                              void* d_out, int out_size, void* d_ws, size_t ws_size,
                              hipStream_t stream) {
    const float* enc        = (const float*)d_in[0];   // [16,2048,1024] f32
    const int*   mask       = (const int*)  d_in[1];   // [16,2048] i32
    const int*   head       = (const int*)  d_in[2];   // [16,32] i32
    const int*   tail       = (const int*)  d_in[3];   // [16,32] i32
    const int*   labels     = (const int*)  d_in[4];   // [16,32] i32
    const float* width_emb  = (const float*)d_in[5];   // [64,8]  f32
    const float* length_emb = (const float*)d_in[6];   // [64,8]  f32
    const float* cls_w      = (const float*)d_in[7];   // [1040,3] f32
    const float* cls_b      = (const float*)d_in[8];   // [3] f32

    float* out = (float*)d_out;                        // 1536 logits + 1 loss

    float* feat    = (float*)d_ws;                                        // [512,1040] f32
    int*   lengths = (int*)((char*)d_ws + (size_t)(B_ * S_) * FEAT * sizeof(float));

    lengths_kernel  <<<B_,        256, 0, stream>>>(mask, lengths);
    span_pool_kernel<<<B_ * S_,   256, 0, stream>>>(enc, head, tail, width_emb,
                                                    length_emb, lengths, feat);
    gemm_wmma_kernel<<<(B_ * S_) / 16, 32, 0, stream>>>(feat, cls_w, cls_b, out);
    loss_kernel     <<<1,         512, 0, stream>>>(out, labels, out + B_ * S_ * NLAB);
}